// GlobalAggregator_81303730913788
// MI455X (gfx1250) — compile-verified
//
#include <hip/hip_runtime.h>
#include <hip/hip_bf16.h>

// MI455X (gfx1250, wave32) implementation.
// Kernel A: attention (feat @ w1 via v_wmma_f32_16x16x4_f32, leaky-relu, @w2,
//           softmax over K=16, weighted aggregation) -> agg in workspace.
// Kernel B: out = relu(concat(self, agg) @ w3) via v_wmma_f32_16x16x4_f32.
//
// Weights are staged in LDS interleaved by k-row pairs with a 288-dword row
// stride: lds[kpair*288 + col*2 + par] = w[(2*kpair+par)*128 + col].
//  - B tile (rows d0,d0+1 / cols 16j..16j+15) = one aligned ds_load_b64,
//    merged pairwise by the compiler into ds_load_2addr_b64.
//  - 288 mod 64 == 32, so odd kpair rows (lanes 16-31) hit banks 32..63 while
//    even kpair rows (lanes 0-15) hit banks 0..31 -> conflict-free.
// Occupancy is pinned at >=2 waves/SIMD via __launch_bounds__ so the serial
// softmax epilogue overlaps across waves; kk loops use a bounded unroll (x4)
// to keep a deep scheduling window without the full-unroll register blowup.

typedef float v2f __attribute__((ext_vector_type(2)));
typedef float v8f __attribute__((ext_vector_type(8)));

#define RB 256
#define NB 100
#define KB 16
#define DB 128
#define ROWS (RB * NB)          // 25600
#define WPSTRIDE 288            // padded dwords per interleaved k-row pair

#define WAVES_A 4
#define GRID_A  1024
#define WAVES_B 4
#define GRID_B  400             // 400*4 = 1600 = ROWS/16 M-tiles

// Kernel A LDS map (floats):
//   [0, 64*288)            interleaved w1 rows 0..127
//   [18432, 18432+128)     w1 row 128 (the concatenated-weight column)
//   [18560, 18560+128)     w2
#define A_W1LAST 18432
#define A_W2     18560
#define A_TOTAL  18688

// ---------------------------------------------------------------------------
// Kernel A: per-wave attention row.  One wave handles one (b,n) row at a time.
// ---------------------------------------------------------------------------
__global__ __launch_bounds__(32 * WAVES_A, 2)
void ga_attn_kernel(const float* __restrict__ nbr,    // [ROWS,16,128]
                    const float* __restrict__ nwgt,   // [ROWS,16]
                    const float* __restrict__ extra,  // [ROWS,128]
                    const float* __restrict__ w1,     // [129,128]
                    const float* __restrict__ w2,     // [128]
                    float* __restrict__ agg,          // [ROWS,128] (workspace)
                    int nslots)
{
    extern __shared__ float smem[];

    // Stage w1 (interleaved+padded) and w2 into LDS.
    for (int i = threadIdx.x; i < 129 * 128; i += blockDim.x) {
        const int krow = i >> 7;
        const int col  = i & 127;
        const float v  = w1[i];
        const int dst  = (krow < 128) ? ((krow >> 1) * WPSTRIDE + col * 2 + (krow & 1))
                                      : (A_W1LAST + col);
        smem[dst] = v;
    }
    for (int i = threadIdx.x; i < 128; i += blockDim.x) smem[A_W2 + i] = w2[i];
    __syncthreads();

    const int lane = threadIdx.x & 31;
    const int wave = threadIdx.x >> 5;
    const int lo   = lane & 15;          // column-within-tile / A-row id
    const int hi   = lane >> 4;          // 0: lanes 0-15, 1: lanes 16-31
    const int slot = blockIdx.x * WAVES_A + wave;

    for (int r = slot; r < ROWS; r += nslots) {
        const float* nbr_r = nbr   + (size_t)r * (KB * DB);
        const float* ex_r  = extra + (size_t)r * DB;
        const float* nw_r  = nwgt  + (size_t)r * KB;

        v8f acc[8];
        #pragma unroll
        for (int j = 0; j < 8; ++j) { v8f z = {}; acc[j] = z; }

        // alpha[16,128] = (extra * nbr)[16,128] @ w1[0:128, :]
        // Software-pipelined over kk: A tile for kk+1 issued before WMMAs of kk.
        v2f nb = *(const v2f*)(nbr_r + lo * DB + 2 * hi);
        v2f ex = *(const v2f*)(ex_r + 2 * hi);
        #pragma unroll 4
        for (int kk = 0; kk < 32; ++kk) {
            v2f a;
            a.x = nb.x * ex.x;
            a.y = nb.y * ex.y;
            if (kk < 31) {                        // prefetch next A tile
                const int d1 = 4 * (kk + 1) + 2 * hi;
                nb = *(const v2f*)(nbr_r + lo * DB + d1);
                ex = *(const v2f*)(ex_r + d1);
            }
            // load all 8 B tiles first (4x ds_load_2addr_b64 in flight)
            const float* bbase = smem + (2 * kk + hi) * WPSTRIDE + lo * 2;
            v2f bt[8];
            #pragma unroll
            for (int j = 0; j < 8; ++j) bt[j] = *(const v2f*)(bbase + 32 * j);
            #pragma unroll
            for (int j = 0; j < 8; ++j) {
                acc[j] = __builtin_amdgcn_wmma_f32_16x16x4_f32(
                    false, a, false, bt[j], (short)0, acc[j], false, false);
            }
        }

        // rank-1 update for the concatenated weight column:
        // alpha[k,d] += nwgt[k] * w1[128, d]
        float nw8[8];
        #pragma unroll
        for (int t = 0; t < 8; ++t) nw8[t] = nw_r[8 * hi + t];
        #pragma unroll
        for (int j = 0; j < 8; ++j) {
            const float w1l = smem[A_W1LAST + 16 * j + lo];
            #pragma unroll
            for (int t = 0; t < 8; ++t) acc[j][t] = fmaf(nw8[t], w1l, acc[j][t]);
        }

        // leaky-relu(0.2) then dot with w2 -> logits p[k], k = t + 8*hi
        float w2l[8];
        #pragma unroll
        for (int j = 0; j < 8; ++j) w2l[j] = smem[A_W2 + 16 * j + lo];
        float p[8];
        #pragma unroll
        for (int t = 0; t < 8; ++t) {
            float s = 0.0f;
            #pragma unroll
            for (int j = 0; j < 8; ++j) {
                float x = acc[j][t];
                x = (x > 0.0f) ? x : 0.2f * x;
                s = fmaf(x, w2l[j], s);
            }
            p[t] = s;
        }
        // sum the 16 per-lane partials within each half-wave
        #pragma unroll
        for (int t = 0; t < 8; ++t) {
            float s = p[t];
            s += __shfl_xor(s, 1, 32);
            s += __shfl_xor(s, 2, 32);
            s += __shfl_xor(s, 4, 32);
            s += __shfl_xor(s, 8, 32);
            p[t] = s;
        }

        // softmax over all 16 k (halves combined via xor-16)
        float m = p[0];
        #pragma unroll
        for (int t = 1; t < 8; ++t) m = fmaxf(m, p[t]);
        m = fmaxf(m, __shfl_xor(m, 16, 32));
        float e[8];
        float s = 0.0f;
        #pragma unroll
        for (int t = 0; t < 8; ++t) { e[t] = __expf(p[t] - m); s += e[t]; }
        s += __shfl_xor(s, 16, 32);
        const float inv = 1.0f / s;
        #pragma unroll
        for (int t = 0; t < 8; ++t) e[t] *= inv;

        // broadcast full alpha[0..15] to every lane
        float al[16];
        #pragma unroll
        for (int t = 0; t < 8; ++t) {
            al[t]     = __shfl(e[t], 0, 32);   // lane 0  holds k = t
            al[t + 8] = __shfl(e[t], 16, 32);  // lane 16 holds k = t+8
        }

        // agg[d] = sum_k alpha[k] * nbr[k,d]; lane covers d = 4*lane..4*lane+3
        float4 a4 = make_float4(0.f, 0.f, 0.f, 0.f);
        #pragma unroll
        for (int k = 0; k < KB; ++k) {
            const float4 nv = *(const float4*)(nbr_r + k * DB + lane * 4);
            a4.x = fmaf(al[k], nv.x, a4.x);
            a4.y = fmaf(al[k], nv.y, a4.y);
            a4.z = fmaf(al[k], nv.z, a4.z);
            a4.w = fmaf(al[k], nv.w, a4.w);
        }
        *(float4*)(agg + (size_t)r * DB + lane * 4) = a4;
    }
}

// ---------------------------------------------------------------------------
// Kernel B: out = relu([self | agg] @ w3), w3: [256,128].
// One wave per 16-row M-tile; 8 N-tiles x 64 K-steps of f32 WMMA.
// ---------------------------------------------------------------------------
__global__ __launch_bounds__(32 * WAVES_B, 2)
void ga_out_kernel(const float* __restrict__ selfv,  // [ROWS,128]
                   const float* __restrict__ agg,    // [ROWS,128]
                   const float* __restrict__ w3,     // [256,128]
                   float* __restrict__ out)          // [ROWS,128]
{
    extern __shared__ float smem[];

    // Stage w3 interleaved by k-row pairs (128 kpairs, padded stride).
    for (int i = threadIdx.x; i < 256 * 128; i += blockDim.x) {
        const int krow = i >> 7;
        const int col  = i & 127;
        smem[(krow >> 1) * WPSTRIDE + col * 2 + (krow & 1)] = w3[i];
    }
    __syncthreads();

    const int lane = threadIdx.x & 31;
    const int wave = threadIdx.x >> 5;
    const int lo   = lane & 15;
    const int hi   = lane >> 4;
    const int ntiles = ROWS / 16;         // 1600

    for (int mt = blockIdx.x * WAVES_B + wave; mt < ntiles;
         mt += gridDim.x * WAVES_B) {
        const int row = mt * 16 + lo;     // A-tile row for this lane
        const float* srow = selfv + (size_t)row * DB;
        const float* arow = agg   + (size_t)row * DB;

        v8f acc[8];
        #pragma unroll
        for (int j = 0; j < 8; ++j) { v8f z = {}; acc[j] = z; }

        // K loop over all 256 input features; src pointer split at k=128.
        v2f a = *(const v2f*)(srow + 2 * hi);
        #pragma unroll 4
        for (int kk = 0; kk < 64; ++kk) {
            const v2f av = a;
            if (kk < 63) {                        // prefetch next A tile
                const int k1 = 4 * (kk + 1) + 2 * hi;
                a = (kk + 1 < 32) ? *(const v2f*)(srow + k1)
                                  : *(const v2f*)(arow + (k1 - 128));
            }
            const float* bbase = smem + (2 * kk + hi) * WPSTRIDE + lo * 2;
            v2f bt[8];
            #pragma unroll
            for (int j = 0; j < 8; ++j) bt[j] = *(const v2f*)(bbase + 32 * j);
            #pragma unroll
            for (int j = 0; j < 8; ++j) {
                acc[j] = __builtin_amdgcn_wmma_f32_16x16x4_f32(
                    false, av, false, bt[j], (short)0, acc[j], false, false);
            }
        }

        // relu + store: C layout element (M = t + 8*hi, N = 16j + lo)
        #pragma unroll
        for (int j = 0; j < 8; ++j) {
            #pragma unroll
            for (int t = 0; t < 8; ++t) {
                const int orow = mt * 16 + t + 8 * hi;
                float v = acc[j][t];
                v = (v > 0.0f) ? v : 0.0f;
                out[(size_t)orow * DB + 16 * j + lo] = v;
            }
        }
    }
}

extern "C" void kernel_launch(void* const* d_in, const int* in_sizes, int n_in,
                              void* d_out, int out_size, void* d_ws, size_t ws_size,
                              hipStream_t stream) {
    (void)in_sizes; (void)n_in; (void)out_size; (void)ws_size;
    const float* selfv = (const float*)d_in[0];  // [B,N,D]
    const float* nbr   = (const float*)d_in[1];  // [B,N,K,D]
    const float* nwgt  = (const float*)d_in[2];  // [B,N,K]
    const float* extra = (const float*)d_in[3];  // [B,N,D]
    const float* w1    = (const float*)d_in[4];  // [129,128]
    const float* w2    = (const float*)d_in[5];  // [128,1]
    const float* w3    = (const float*)d_in[6];  // [256,128]
    float* out = (float*)d_out;                  // [B,N,D]
    float* agg = (float*)d_ws;                   // [ROWS,128] scratch (13.1 MB)

    const size_t ldsA = (size_t)A_TOTAL * sizeof(float);            // ~74.8 KB
    ga_attn_kernel<<<dim3(GRID_A), dim3(32 * WAVES_A), ldsA, stream>>>(
        nbr, nwgt, extra, w1, w2, agg, GRID_A * WAVES_A);

    const size_t ldsB = (size_t)(128 * WPSTRIDE) * sizeof(float);   // 144 KB
    ga_out_kernel<<<dim3(GRID_B), dim3(32 * WAVES_B), ldsB, stream>>>(
        selfv, agg, w3, out);
}